// SelfAttentionLayer_66245575573907
// MI455X (gfx1250) — compile-verified
//
#include <hip/hip_runtime.h>
#include <math.h>

typedef __attribute__((ext_vector_type(2))) float v2f;
typedef __attribute__((ext_vector_type(8))) float v8f;
typedef __attribute__((ext_vector_type(2))) int   v2i;

#define DIM   8192
#define NROW  200
#define NPAD  208           // 13 * 16
#define KB    32
#define XSTR  36            // LDS row stride (floats): mult of 4, avoids bank conflicts
#define PSTR  212           // LDS stride for p tile in out kernel
#define CC    90.50966799187809f   // sqrt(8192)
#define EPS   1e-5f

#define BUFSZ ((NPAD + 128) * XSTR)   // floats per double-buffer stage (12096)

// ---- CDNA5 async global->LDS (ASYNCcnt) with sync fallback --------------
#if __has_builtin(__builtin_amdgcn_global_load_async_to_lds_b64)
#define ASYNC_CAPABLE 1
#define ASYNC_B64(g, l)                                                     \
    __builtin_amdgcn_global_load_async_to_lds_b64(                          \
        (__attribute__((address_space(1))) v2i*)(g),                        \
        (__attribute__((address_space(3))) v2i*)(l), 0, 0)
#else
#define ASYNC_CAPABLE 0
#endif

#if __has_builtin(__builtin_amdgcn_s_wait_asynccnt)
#define WAIT_ASYNC(n) __builtin_amdgcn_s_wait_asynccnt(n)
#else
#define WAIT_ASYNC(n) asm volatile("s_wait_asynccnt %0" ::"n"(n) : "memory")
#endif

// ---------------------------------------------------------------------------
// Kernel 1: q/k/v = x @ W.T + b   (NT GEMM, K = 8192, weight-stream bound)
// grid: (64, 3)  block: 256 (8 waves). Block = 128 output cols x 208 rows.
// Double-buffered LDS pipeline: async copies for chunk i+1 overlap WMMA on i.
// ---------------------------------------------------------------------------
__global__ void __launch_bounds__(256) qkv_kernel(
    const float* __restrict__ x,
    const float* __restrict__ Wq, const float* __restrict__ bq,
    const float* __restrict__ Wk, const float* __restrict__ bk,
    const float* __restrict__ Wv, const float* __restrict__ bv,
    float* __restrict__ qp, float* __restrict__ kp, float* __restrict__ vp)
{
    const int wsel = blockIdx.y;
    const float* __restrict__ W    = (wsel == 0) ? Wq : (wsel == 1) ? Wk : Wv;
    const float* __restrict__ bias = (wsel == 0) ? bq : (wsel == 1) ? bk : bv;
    float* __restrict__ out        = (wsel == 0) ? qp : (wsel == 1) ? kp : vp;

    const int o0 = blockIdx.x * 128;

    extern __shared__ float lds[];     // 2 * BUFSZ floats (96768 B dynamic)

    const int tid  = threadIdx.x;
    const int wave = tid >> 5;
    const int lane = tid & 31;
    const int lm   = lane & 15;
    const int hi   = lane >> 4;

    // Pre-zero pad rows (200..207) of the x slab in BOTH buffers, once.
    // 8 rows * 32 cols = 256 floats per buffer: one element per thread.
    {
        const int row = 200 + (tid >> 5), c = tid & 31;
        lds[row * XSTR + c]         = 0.f;
        lds[BUFSZ + row * XSTR + c] = 0.f;
    }
    __syncthreads();

    auto issue = [&](int buf, int k0) {
        float* xs = lds + buf * BUFSZ;
        float* wl = xs + NPAD * XSTR;
#if ASYNC_CAPABLE
        // x slab: 200 real rows x 16 float2 chunks (pad rows pre-zeroed).
        // Waves 0-3 issue 13 loads/thread, waves 4-7 issue 12.
        for (int t = tid; t < NPAD * 16; t += 256) {
            const int row = t >> 4, c2 = t & 15;
            if (row < NROW)
                ASYNC_B64(x + (size_t)row * DIM + k0 + c2 * 2,
                          xs + row * XSTR + c2 * 2);
        }
        // W slab: 128 rows x 16 float2 chunks -> 8 loads/thread (uniform).
        for (int t = tid; t < 128 * 16; t += 256) {
            const int row = t >> 4, c2 = t & 15;
            ASYNC_B64(W + (size_t)(o0 + row) * DIM + k0 + c2 * 2,
                      wl + row * XSTR + c2 * 2);
        }
#else
        for (int t = tid; t < NPAD * 8; t += 256) {
            const int row = t >> 3, c4 = t & 7;
            if (row < NROW)
                *(float4*)(&xs[row * XSTR + c4 * 4]) =
                    *(const float4*)(x + (size_t)row * DIM + k0 + c4 * 4);
        }
        for (int t = tid; t < 128 * 8; t += 256) {
            const int row = t >> 3, c4 = t & 7;
            *(float4*)(&wl[row * XSTR + c4 * 4]) =
                *(const float4*)(W + (size_t)(o0 + row) * DIM + k0 + c4 * 4);
        }
#endif
    };

    v8f zero = {};
    v8f acc[13];
#pragma unroll
    for (int i = 0; i < 13; ++i) acc[i] = zero;

    const int NITER = DIM / KB;        // 256
    issue(0, 0);

    for (int it = 0; it < NITER; ++it) {
        const int buf = it & 1;
        if (it + 1 < NITER) issue(buf ^ 1, (it + 1) * KB);
#if ASYNC_CAPABLE
        // Wait until only the just-issued batch is outstanding -> buffer
        // `buf` (previous batch) is fully in LDS. Counts are per-wave.
        if (it + 1 < NITER) {
            if (wave < 4) WAIT_ASYNC(21);   // 13 x-loads + 8 W-loads
            else          WAIT_ASYNC(20);   // 12 x-loads + 8 W-loads
        } else {
            WAIT_ASYNC(0);
        }
#endif
        __syncthreads();                   // whole buffer visible to all waves

        const float* xs = lds + buf * BUFSZ;
        const float* wl = xs + NPAD * XSTR;
#pragma unroll
        for (int kk = 0; kk < KB; kk += 4) {
            v2f bfrag = *(const v2f*)(&wl[(wave * 16 + lm) * XSTR + kk + 2 * hi]);
#pragma unroll
            for (int mt = 0; mt < 13; ++mt) {
                v2f afrag = *(const v2f*)(&xs[(mt * 16 + lm) * XSTR + kk + 2 * hi]);
                acc[mt] = __builtin_amdgcn_wmma_f32_16x16x4_f32(
                    false, afrag, false, bfrag, (short)0, acc[mt], false, false);
            }
        }
        __syncthreads();                   // done reading before next overwrite
    }

    const int o = o0 + wave * 16 + lm;
    const float bval = bias[o];
#pragma unroll
    for (int mt = 0; mt < 13; ++mt) {
#pragma unroll
        for (int r = 0; r < 8; ++r) {
            const int row = mt * 16 + r + 8 * hi;   // <= 207, pad rows = bias (masked later)
            out[(size_t)row * DIM + o] = acc[mt][r] + bval;
        }
    }
}

// ---------------------------------------------------------------------------
// Kernel 2: s = q @ k.T  (208x208 padded, K = 8192). grid 169, block 32.
// ---------------------------------------------------------------------------
__global__ void __launch_bounds__(32) score_kernel(
    const float* __restrict__ qp, const float* __restrict__ kp,
    float* __restrict__ sp)
{
    const int mi = blockIdx.x % 13;
    const int mj = blockIdx.x / 13;

    __shared__ __align__(16) float qs[16 * XSTR];
    __shared__ __align__(16) float ks[16 * XSTR];

    const int lane = threadIdx.x;
    const int lm = lane & 15, hi = lane >> 4;

    v8f acc = {};

    for (int k0 = 0; k0 < DIM; k0 += KB) {
        __syncthreads();
        for (int t = lane; t < 16 * 8; t += 32) {
            const int row = t >> 3, c4 = t & 7;
            *(float4*)(&qs[row * XSTR + c4 * 4]) =
                *(const float4*)(qp + (size_t)(mi * 16 + row) * DIM + k0 + c4 * 4);
            *(float4*)(&ks[row * XSTR + c4 * 4]) =
                *(const float4*)(kp + (size_t)(mj * 16 + row) * DIM + k0 + c4 * 4);
        }
        __syncthreads();
#pragma unroll
        for (int kk = 0; kk < KB; kk += 4) {
            v2f a = *(const v2f*)(&qs[lm * XSTR + kk + 2 * hi]);
            v2f b = *(const v2f*)(&ks[lm * XSTR + kk + 2 * hi]);
            acc = __builtin_amdgcn_wmma_f32_16x16x4_f32(
                false, a, false, b, (short)0, acc, false, false);
        }
    }
#pragma unroll
    for (int r = 0; r < 8; ++r) {
        const int row = mi * 16 + r + 8 * hi;
        sp[(size_t)row * NPAD + mj * 16 + lm] = acc[r];
    }
}

// ---------------------------------------------------------------------------
// Kernel 3: per-row layernorm (unbiased var) -> sin -> softmax(/sqrt(D)).
// In-place on padded s; zeroes pad rows/cols. grid 208, block 256.
// ---------------------------------------------------------------------------
__global__ void __launch_bounds__(256) norm_softmax_kernel(float* __restrict__ sp)
{
    const int i   = blockIdx.x;
    const int tid = threadIdx.x;
    float* row = sp + (size_t)i * NPAD;

    if (i >= NROW) {                       // pad rows -> 0
        if (tid < NPAD) row[tid] = 0.f;
        return;
    }

    __shared__ float red[256];
    const float val = (tid < NROW) ? row[tid] : 0.f;

    // mean
    red[tid] = val; __syncthreads();
    for (int s = 128; s > 0; s >>= 1) { if (tid < s) red[tid] += red[tid + s]; __syncthreads(); }
    const float mean = red[0] * (1.0f / NROW);
    __syncthreads();

    // unbiased variance
    const float d = (tid < NROW) ? (val - mean) : 0.f;
    red[tid] = d * d; __syncthreads();
    for (int s = 128; s > 0; s >>= 1) { if (tid < s) red[tid] += red[tid + s]; __syncthreads(); }
    const float var = red[0] * (1.0f / (NROW - 1));
    __syncthreads();

    const float rstd = rsqrtf(var + EPS);
    const float z = (tid < NROW) ? (__sinf(d * rstd) * (1.0f / CC)) : -INFINITY;

    // max
    red[tid] = z; __syncthreads();
    for (int s = 128; s > 0; s >>= 1) { if (tid < s) red[tid] = fmaxf(red[tid], red[tid + s]); __syncthreads(); }
    const float m = red[0];
    __syncthreads();

    const float e = (tid < NROW) ? __expf(z - m) : 0.f;
    red[tid] = e; __syncthreads();
    for (int s = 128; s > 0; s >>= 1) { if (tid < s) red[tid] += red[tid + s]; __syncthreads(); }
    const float inv = 1.0f / red[0];

    if (tid < NROW)      row[tid] = e * inv;
    else if (tid < NPAD) row[tid] = 0.f;   // pad cols -> 0
}

// ---------------------------------------------------------------------------
// Kernel 4: out = p @ v  (NN GEMM, K = 208). grid (64, 13), block 256.
// ---------------------------------------------------------------------------
__global__ void __launch_bounds__(256) out_kernel(
    const float* __restrict__ sp, const float* __restrict__ vp,
    float* __restrict__ out)
{
    const int o0 = blockIdx.x * 128;
    const int m0 = blockIdx.y * 16;

    __shared__ __align__(16) float ps[16 * PSTR];   // 13568 B

    const int tid  = threadIdx.x;
    const int wave = tid >> 5;
    const int lane = tid & 31;
    const int lm = lane & 15, hi = lane >> 4;

    // Stage the 16x208 p tile: 16 rows x 52 float4
    for (int t = tid; t < 16 * 52; t += 256) {
        const int row = t / 52, c4 = t % 52;
        *(float4*)(&ps[row * PSTR + c4 * 4]) =
            *(const float4*)(sp + (size_t)(m0 + row) * NPAD + c4 * 4);
    }
    __syncthreads();

    const int o = o0 + wave * 16 + lm;
    v8f acc = {};
#pragma unroll
    for (int kk = 0; kk < NPAD; kk += 4) {
        v2f a = *(const v2f*)(&ps[lm * PSTR + kk + 2 * hi]);
        v2f b;
        b.x = vp[(size_t)(kk + 2 * hi) * DIM + o];       // coalesced across lanes
        b.y = vp[(size_t)(kk + 1 + 2 * hi) * DIM + o];
        acc = __builtin_amdgcn_wmma_f32_16x16x4_f32(
            false, a, false, b, (short)0, acc, false, false);
    }
#pragma unroll
    for (int r = 0; r < 8; ++r) {
        const int row = m0 + r + 8 * hi;
        if (row < NROW)
            out[(size_t)row * DIM + o] = acc[r];
    }
}

// ---------------------------------------------------------------------------
extern "C" void kernel_launch(void* const* d_in, const int* in_sizes, int n_in,
                              void* d_out, int out_size, void* d_ws, size_t ws_size,
                              hipStream_t stream) {
    (void)in_sizes; (void)n_in; (void)out_size; (void)ws_size;

    const float* x  = (const float*)d_in[0];
    const float* Wq = (const float*)d_in[1];
    const float* bq = (const float*)d_in[2];
    const float* Wk = (const float*)d_in[3];
    const float* bk = (const float*)d_in[4];
    const float* Wv = (const float*)d_in[5];
    const float* bv = (const float*)d_in[6];
    float* out = (float*)d_out;

    float* ws = (float*)d_ws;
    const size_t qkv_elems = (size_t)NPAD * DIM;   // 1,703,936 floats each
    float* qp = ws;
    float* kp = ws + qkv_elems;
    float* vp = ws + 2 * qkv_elems;
    float* sp = ws + 3 * qkv_elems;                // 208*208 floats

    dim3 g1(DIM / 128, 3);
    const size_t lds_bytes = 2 * BUFSZ * sizeof(float);   // 96768 B
    qkv_kernel<<<g1, 256, lds_bytes, stream>>>(x, Wq, bq, Wk, bk, Wv, bv, qp, kp, vp);

    score_kernel<<<13 * 13, 32, 0, stream>>>(qp, kp, sp);

    norm_softmax_kernel<<<NPAD, 256, 0, stream>>>(sp);

    dim3 g4(DIM / 128, 13);
    out_kernel<<<g4, 256, 0, stream>>>(sp, vp, out);
}